// Decoder_29059748725688
// MI455X (gfx1250) — compile-verified
//
#include <hip/hip_runtime.h>

typedef float v2f __attribute__((ext_vector_type(2)));
typedef float v8f __attribute__((ext_vector_type(8)));

#define LN_EPS 1e-5f

// =====================================================================
// WMMA fp32 GEMM:  C[M,N] = op(A[M,K] @ B + bias)
// One 16x16 output tile per wave32, V_WMMA_F32_16X16X4_F32, K stepped by 4.
// A  : 16x4 frag  -> lanes 0-15 rows, VGPR0/1 = K{0,1} (lanes<16) / K{2,3}
// B  : 4x16 frag  -> mirrored layout (rows K, cols across lanes)
// C/D: 16x16 f32  -> 8 VGPRs, lane cols, M = r (+8 for lanes>=16)
// BT: B indexed transposed (B[n,k])   NG: guard N edge (final N=13)
// =====================================================================
template<bool BT, bool BIAS, bool RELU, bool NG>
__global__ void gemm_wmma_f32(const float* __restrict__ A, const float* __restrict__ B,
                              const float* __restrict__ bias, float* __restrict__ C,
                              int M, int N, int K, int lda, int ldb, int ldc, int cOff)
{
  const int wave = threadIdx.x >> 5;
  const int lane = threadIdx.x & 31;
  const int mt   = blockIdx.y * (blockDim.x >> 5) + wave;
  const int m0   = mt << 4;
  const int n0   = blockIdx.x << 4;
  if (m0 >= M) return;                       // wave-uniform exit (EXEC stays all-1s)
  const bool hi   = lane >= 16;
  const int  l16  = lane & 15;
  const int  arow = m0 + l16;
  const int  ncol = n0 + l16;
  const bool nok  = (!NG) || (ncol < N);

  v8f acc = {0.f, 0.f, 0.f, 0.f, 0.f, 0.f, 0.f, 0.f};
  const int kb = hi ? 2 : 0;
  for (int k = 0; k < K; k += 4) {
    const int ka = k + kb;
    v2f a, b;
    a.x = A[arow * lda + ka];
    a.y = A[arow * lda + ka + 1];
    if (BT) {
      b.x = nok ? B[ncol * ldb + ka]       : 0.f;   // contiguous b64 pair
      b.y = nok ? B[ncol * ldb + ka + 1]   : 0.f;
    } else {
      b.x = nok ? B[(ka    ) * ldb + ncol] : 0.f;
      b.y = nok ? B[(ka + 1) * ldb + ncol] : 0.f;
    }
    acc = __builtin_amdgcn_wmma_f32_16x16x4_f32(false, a, false, b, (short)0, acc,
                                                false, false);
  }
  float bv = 0.f;
  if (BIAS && nok) bv = bias[ncol];
  if (nok) {
#pragma unroll
    for (int r = 0; r < 8; ++r) {
      float v = acc[r] + bv;
      if (RELU) v = fmaxf(v, 0.f);
      C[(size_t)(m0 + r + (hi ? 8 : 0)) * ldc + cOff + ncol] = v;
    }
  }
}

// =====================================================================
// KNN: per-thread top-16 smallest squared distances, LDS-tiled coarse set.
// =====================================================================
__global__ void knn_kernel(const float* __restrict__ fp, const float* __restrict__ cp,
                           int Nf, int Nc, int* __restrict__ idx)
{
  __shared__ float sx[256], sy[256], sz[256];
  const int t = threadIdx.x;
  const int n = blockIdx.x * 256 + t;
  float fx = 0.f, fy = 0.f, fz = 0.f;
  if (n < Nf) { fx = fp[n * 3]; fy = fp[n * 3 + 1]; fz = fp[n * 3 + 2]; }
  float bd[16]; int bi[16];
#pragma unroll
  for (int k = 0; k < 16; ++k) { bd[k] = 3.4e38f; bi[k] = 0; }

  for (int base = 0; base < Nc; base += 256) {
    __syncthreads();
    if (base + t < Nc) {
      sx[t] = cp[(base + t) * 3];
      sy[t] = cp[(base + t) * 3 + 1];
      sz[t] = cp[(base + t) * 3 + 2];
    }
    if (base + 256 + t < Nc)                           // gfx1250 global_prefetch_b8
      __builtin_prefetch(&cp[(base + 256 + t) * 3], 0, 1);
    __syncthreads();
    const int cnt = (Nc - base < 256) ? (Nc - base) : 256;
    for (int j = 0; j < cnt; ++j) {
      const float dx = fx - sx[j], dy = fy - sy[j], dz = fz - sz[j];
      float d = dx * dx + dy * dy + dz * dz;
      if (d < bd[15]) {
        int ii = base + j;
#pragma unroll
        for (int k = 0; k < 16; ++k) {                 // bubble-insert, keeps ascending
          if (d < bd[k]) {
            float td = bd[k]; bd[k] = d; d = td;
            int   ti = bi[k]; bi[k] = ii; ii = ti;
          }
        }
      }
    }
  }
  if (n < Nf) {
#pragma unroll
    for (int k = 0; k < 16; ++k) idx[n * 16 + k] = bi[k];
  }
}

// =====================================================================
// Attention (wave per fine point):
//   s_k = (q Wk^T) . x[idx_k]  ->  softmax over 16  ->  z = sum w_k x[idx_k]
// (q.bk constant over k -> dropped; bv folded into the z@Wv GEMM)
// =====================================================================
template<int DD>
__global__ void attn_kernel(const float* __restrict__ qt, const float* __restrict__ xc,
                            const int* __restrict__ idx, float* __restrict__ z, int Nf)
{
  constexpr int E = DD / 32;
  const int wave = threadIdx.x >> 5, lane = threadIdx.x & 31;
  const int n = blockIdx.x * (blockDim.x >> 5) + wave;
  if (n >= Nf) return;
  float ql[E];
#pragma unroll
  for (int e = 0; e < E; ++e) ql[e] = qt[(size_t)n * DD + e * 32 + lane];
  int nid[16];
#pragma unroll
  for (int k = 0; k < 16; ++k) nid[k] = idx[n * 16 + k];
  float sc[16];
#pragma unroll
  for (int k = 0; k < 16; ++k) {
    const float* row = xc + (size_t)nid[k] * DD;
    float p = 0.f;
#pragma unroll
    for (int e = 0; e < E; ++e) p += ql[e] * row[e * 32 + lane];
#pragma unroll
    for (int o = 16; o > 0; o >>= 1) p += __shfl_xor(p, o, 32);  // wave32 reduce
    sc[k] = p;
  }
  const float scale = rsqrtf((float)DD);
  float m = -3.4e38f;
#pragma unroll
  for (int k = 0; k < 16; ++k) { sc[k] *= scale; m = fmaxf(m, sc[k]); }
  float s = 0.f;
#pragma unroll
  for (int k = 0; k < 16; ++k) { sc[k] = __expf(sc[k] - m); s += sc[k]; }
  const float inv = 1.f / s;
#pragma unroll
  for (int e = 0; e < E; ++e) {
    float a = 0.f;
#pragma unroll
    for (int k = 0; k < 16; ++k) a += sc[k] * xc[(size_t)nid[k] * DD + e * 32 + lane];
    z[(size_t)n * DD + e * 32 + lane] = a * inv;
  }
}

// =====================================================================
// LayerNorm + residual:  out = LN(h)*g + b + skip   (wave per row)
// =====================================================================
template<int DE>
__global__ void ln_res_kernel(const float* __restrict__ h, const float* __restrict__ sk,
                              const float* __restrict__ g, const float* __restrict__ b,
                              float* __restrict__ out, int Nf)
{
  constexpr int E = DE / 32;
  const int wave = threadIdx.x >> 5, lane = threadIdx.x & 31;
  const int n = blockIdx.x * (blockDim.x >> 5) + wave;
  if (n >= Nf) return;
  float v[E];
  float s = 0.f;
#pragma unroll
  for (int e = 0; e < E; ++e) { v[e] = h[(size_t)n * DE + e * 32 + lane]; s += v[e]; }
#pragma unroll
  for (int o = 16; o > 0; o >>= 1) s += __shfl_xor(s, o, 32);
  const float mu = s / DE;
  float q = 0.f;
#pragma unroll
  for (int e = 0; e < E; ++e) { const float d = v[e] - mu; q += d * d; }
#pragma unroll
  for (int o = 16; o > 0; o >>= 1) q += __shfl_xor(q, o, 32);
  const float inv = rsqrtf(q / DE + LN_EPS);
#pragma unroll
  for (int e = 0; e < E; ++e) {
    const int c = e * 32 + lane;
    out[(size_t)n * DE + c] = (v[e] - mu) * inv * g[c] + b[c] + sk[(size_t)n * DE + c];
  }
}

__global__ void copy_cols_kernel(const float* __restrict__ src, float* __restrict__ dst,
                                 int Nf, int De, int ldc, int off)
{
  const int i = blockIdx.x * blockDim.x + threadIdx.x;
  if (i < Nf * De) {
    const int r = i / De, c = i - r * De;
    dst[(size_t)r * ldc + off + c] = src[i];
  }
}

__global__ void copy_i32_kernel(const int* __restrict__ s, int* __restrict__ d, int n)
{
  const int i = blockIdx.x * blockDim.x + threadIdx.x;
  if (i < n) d[i] = s[i];
}

// =====================================================================
// Host orchestration
// =====================================================================
struct StageP {
  const float *kb, *kw, *lnb, *lng, *m1b, *m1w, *m2b, *m2w, *qb, *qw, *skb, *skw, *vb, *vw;
};

static inline dim3 gemm_grid(int M, int N) { return dim3((N + 15) / 16, (M / 16 + 7) / 8); }

extern "C" void kernel_launch(void* const* d_in, const int* in_sizes, int n_in,
                              void* d_out, int out_size, void* d_ws, size_t ws_size,
                              hipStream_t stream) {
  (void)out_size; (void)ws_size;
  const int DIMS_[4] = {64, 128, 256, 512};
  const int NSZ_[4]  = {32768, 8192, 2048, 512};

  const float* feats[4] = {(const float*)d_in[0], (const float*)d_in[3],
                           (const float*)d_in[6], (const float*)d_in[9]};
  const float* poss[4]  = {(const float*)d_in[1], (const float*)d_in[4],
                           (const float*)d_in[7], (const float*)d_in[10]};
  const int*   lbl0     = (const int*)d_in[2];

  StageP st[3];
  const float *finw = nullptr, *finb = nullptr;

  if (n_in >= 56) {
    if (in_sizes[12] == 13) {
      // jax tree-flatten order (sorted dict keys): final first, then stages
      finb = (const float*)d_in[12];
      finw = (const float*)d_in[13];
      int i = 14;
      for (int s = 0; s < 3; ++s) {
        st[s].kb  = (const float*)d_in[i++]; st[s].kw  = (const float*)d_in[i++];
        st[s].lnb = (const float*)d_in[i++]; st[s].lng = (const float*)d_in[i++];
        st[s].m1b = (const float*)d_in[i++]; st[s].m1w = (const float*)d_in[i++];
        st[s].m2b = (const float*)d_in[i++]; st[s].m2w = (const float*)d_in[i++];
        st[s].qb  = (const float*)d_in[i++]; st[s].qw  = (const float*)d_in[i++];
        st[s].skb = (const float*)d_in[i++]; st[s].skw = (const float*)d_in[i++];
        st[s].vb  = (const float*)d_in[i++]; st[s].vw  = (const float*)d_in[i++];
      }
    } else {
      // dict-insertion order: stages first (w before b), final last
      int i = 12;
      for (int s = 0; s < 3; ++s) {
        st[s].qw  = (const float*)d_in[i++]; st[s].qb  = (const float*)d_in[i++];
        st[s].kw  = (const float*)d_in[i++]; st[s].kb  = (const float*)d_in[i++];
        st[s].vw  = (const float*)d_in[i++]; st[s].vb  = (const float*)d_in[i++];
        st[s].m1w = (const float*)d_in[i++]; st[s].m1b = (const float*)d_in[i++];
        st[s].m2w = (const float*)d_in[i++]; st[s].m2b = (const float*)d_in[i++];
        st[s].skw = (const float*)d_in[i++]; st[s].skb = (const float*)d_in[i++];
        st[s].lng = (const float*)d_in[i++]; st[s].lnb = (const float*)d_in[i++];
      }
      finw = (const float*)d_in[i++];
      finb = (const float*)d_in[i++];
    }
  } else {
    // single flattened blob, sorted-key leaf order
    const float* p = (const float*)d_in[n_in - 1];
    finb = p; p += 13;
    finw = p; p += 64 * 13;
    for (int s = 0; s < 3; ++s) {
      const int Dd = DIMS_[3 - s], De = DIMS_[2 - s];
      st[s].kb  = p; p += Dd;            st[s].kw  = p; p += Dd * Dd;
      st[s].lnb = p; p += De;            st[s].lng = p; p += De;
      st[s].m1b = p; p += De;            st[s].m1w = p; p += (Dd + De) * De;
      st[s].m2b = p; p += De;            st[s].m2w = p; p += De * De;
      st[s].qb  = p; p += Dd;            st[s].qw  = p; p += De * Dd;
      st[s].skb = p; p += De;            st[s].skw = p; p += De * De;
      st[s].vb  = p; p += Dd;            st[s].vw  = p; p += Dd * Dd;
    }
  }

  // workspace layout (bytes)
  char* w = (char*)d_ws;
  int*   idxb  = (int*)  (w + (size_t)0);            //  2 MB  (Nf*16 int)
  float* bufB  = (float*)(w + ((size_t)4  << 20));   // 16 MB  q -> z -> h
  float* bufC  = (float*)(w + ((size_t)20 << 20));   // 16 MB  qt -> h1
  float* comb  = (float*)(w + ((size_t)36 << 20));   // 24 MB  [agg | ef]
  float* skipb = (float*)(w + ((size_t)60 << 20));   //  8 MB
  float* X0    = (float*)(w + ((size_t)68 << 20));   //  8 MB  stage outputs (ping)
  float* X1    = (float*)(w + ((size_t)76 << 20));   //  8 MB  (pong)

  const dim3 blk(256);
  const float* x = feats[3];                         // coarse features [512, 512]

  for (int s = 0; s < 3; ++s) {
    const int Dd = DIMS_[3 - s], De = DIMS_[2 - s];
    const int Nf = NSZ_[2 - s],  Nc = NSZ_[3 - s];
    const float* ef = feats[2 - s];
    const float* ep = poss[2 - s];
    const float* cp = poss[3 - s];
    const StageP& P = st[s];
    float* xout = (s & 1) ? X1 : X0;
    const int waves_grid = (Nf + 7) / 8;

    // 1) KNN indices
    knn_kernel<<<dim3((Nf + 255) / 256), blk, 0, stream>>>(ep, cp, Nf, Nc, idxb);

    // 2) q = ef @ Wq + bq                        [Nf, Dd]
    gemm_wmma_f32<false, true, false, false><<<gemm_grid(Nf, Dd), blk, 0, stream>>>(
        ef, P.qw, P.qb, bufB, Nf, Dd, De, De, Dd, Dd, 0);

    // 3) qt = q @ Wk^T                           [Nf, Dd]
    gemm_wmma_f32<true, false, false, false><<<gemm_grid(Nf, Dd), blk, 0, stream>>>(
        bufB, P.kw, nullptr, bufC, Nf, Dd, Dd, Dd, Dd, Dd, 0);

    // 4) attention: softmax + neighbor blend -> z [Nf, Dd] (overwrites q)
    if (Dd == 512)
      attn_kernel<512><<<dim3(waves_grid), blk, 0, stream>>>(bufC, x, idxb, bufB, Nf);
    else if (Dd == 256)
      attn_kernel<256><<<dim3(waves_grid), blk, 0, stream>>>(bufC, x, idxb, bufB, Nf);
    else
      attn_kernel<128><<<dim3(waves_grid), blk, 0, stream>>>(bufC, x, idxb, bufB, Nf);

    // 5) agg = z @ Wv + bv  -> comb[:, 0:Dd]
    gemm_wmma_f32<false, true, false, false><<<gemm_grid(Nf, Dd), blk, 0, stream>>>(
        bufB, P.vw, P.vb, comb, Nf, Dd, Dd, Dd, Dd, Dd + De, 0);

    // 6) comb[:, Dd:Dd+De] = ef
    copy_cols_kernel<<<dim3((Nf * De + 255) / 256), blk, 0, stream>>>(
        ef, comb, Nf, De, Dd + De, Dd);

    // 7) h1 = relu(comb @ W1 + b1)               [Nf, De] (overwrites qt)
    gemm_wmma_f32<false, true, true, false><<<gemm_grid(Nf, De), blk, 0, stream>>>(
        comb, P.m1w, P.m1b, bufC, Nf, De, Dd + De, Dd + De, De, De, 0);

    // 8) h = h1 @ W2 + b2                        [Nf, De] (overwrites z)
    gemm_wmma_f32<false, true, false, false><<<gemm_grid(Nf, De), blk, 0, stream>>>(
        bufC, P.m2w, P.m2b, bufB, Nf, De, De, De, De, De, 0);

    // 9) skip = relu(ef @ Wskip + bskip)         [Nf, De]
    gemm_wmma_f32<false, true, true, false><<<gemm_grid(Nf, De), blk, 0, stream>>>(
        ef, P.skw, P.skb, skipb, Nf, De, De, De, De, De, 0);

    // 10) x_out = LN(h)*g + b + skip
    if (De == 256)
      ln_res_kernel<256><<<dim3(waves_grid), blk, 0, stream>>>(bufB, skipb, P.lng, P.lnb, xout, Nf);
    else if (De == 128)
      ln_res_kernel<128><<<dim3(waves_grid), blk, 0, stream>>>(bufB, skipb, P.lng, P.lnb, xout, Nf);
    else
      ln_res_kernel<64><<<dim3(waves_grid), blk, 0, stream>>>(bufB, skipb, P.lng, P.lnb, xout, Nf);

    x = xout;
  }

  // final logits = x @ Wf + bf   [32768, 13] -> d_out (N-edge guarded)
  gemm_wmma_f32<false, true, false, true><<<gemm_grid(32768, 13), blk, 0, stream>>>(
      x, finw, finb, (float*)d_out, 32768, 13, 64, 64, 13, 13, 0);

  // second tuple output: lbl0 pass-through (bitwise)
  copy_i32_kernel<<<dim3((32768 + 255) / 256), blk, 0, stream>>>(
      lbl0, ((int*)d_out) + 32768 * 13, 32768);
}